// SoftSplat_30459908063917
// MI455X (gfx1250) — compile-verified
//
#include <hip/hip_runtime.h>
#include <math.h>

// Softmax splatting (forward bilinear scatter) for MI455X / gfx1250.
// out = splat(x * exp(metric)) / splat(exp(metric)), guarded by EPS.
// Atomic/memory bound: ~1.8 GB moved -> ~77us floor at 23.3 TB/s.

#define SPLAT_EPS 1e-7f

namespace {
constexpr int B  = 4;
constexpr int C  = 35;
constexpr int H  = 544;
constexpr int W  = 960;
constexpr int HW = H * W;                        // 522240 (divisible by 4)
constexpr int K  = C + 1;                        // 36 planes per batch (num + den)
constexpr unsigned N_OUT  = (unsigned)B * C * HW;    // 73,113,600
constexpr unsigned N_DEN  = (unsigned)B * HW;        //  2,088,960
constexpr unsigned N_OUT4 = N_OUT / 4;
constexpr unsigned N_DEN4 = N_DEN / 4;
constexpr unsigned HW4    = (unsigned)HW / 4;        // 130560
}

// ---------------------------------------------------------------------------
// Kernel 1: zero the accumulators with b128 stores (pure bandwidth pass).
// Numerators accumulate directly in d_out; denominator plane in d_ws.
// ---------------------------------------------------------------------------
__global__ void softsplat_zero(float4* __restrict__ out4, float4* __restrict__ den4) {
    const float4 z = make_float4(0.f, 0.f, 0.f, 0.f);
    const unsigned stride = gridDim.x * blockDim.x;
    for (unsigned j = blockIdx.x * blockDim.x + threadIdx.x; j < N_OUT4; j += stride)
        out4[j] = z;
    for (unsigned j = blockIdx.x * blockDim.x + threadIdx.x; j < N_DEN4; j += stride)
        den4[j] = z;
}

// ---------------------------------------------------------------------------
// Kernel 2: scatter. blockIdx.y = b*K + c  (144 planes); only the pixel index
// runs in a grid-stride loop, so all plane bases are wave-uniform SGPRs and
// the per-iteration address math is base + p. Channel c == C scatters the
// denominator (val = exp(metric)). A wave32 covers 32 consecutive source
// pixels of one plane: coalesced loads, spatially clustered atomic targets.
// ---------------------------------------------------------------------------
__global__ void softsplat_scatter(const float* __restrict__ x,
                                  const float* __restrict__ flow,
                                  const float* __restrict__ metric,
                                  float* __restrict__ out,
                                  float* __restrict__ den) {
    const unsigned z = blockIdx.y;            // b*K + c
    const unsigned b = z / (unsigned)K;
    const unsigned c = z - b * (unsigned)K;
    const bool isNum = (c < (unsigned)C);

    // b*C + c == z - b
    const float* __restrict__ xplane = x + (size_t)(z - b) * HW;
    float* __restrict__ plane = isNum ? (out + (size_t)(z - b) * HW)
                                      : (den + (size_t)b * HW);
    const float* __restrict__ fxp = flow + ((size_t)b * 2 + 0) * HW;
    const float* __restrict__ fyp = flow + ((size_t)b * 2 + 1) * HW;
    const float* __restrict__ mp  = metric + (size_t)b * HW;

    const unsigned stride = gridDim.x * blockDim.x;
    for (unsigned p = blockIdx.x * blockDim.x + threadIdx.x; p < (unsigned)HW; p += stride) {
        // Stream next iteration's x element toward the caches.
        const unsigned np = p + stride;
        if (isNum && np < (unsigned)HW)
            __builtin_prefetch(xplane + np, 0, 1);

        // flow/metric (25 MB) are reused by all 36 channels -> L2-resident.
        const float dx = fxp[p];
        const float dy = fyp[p];
        const float m  = expf(mp[p]);
        const float val = isNum ? xplane[p] * m : m;

        const int   px  = (int)(p % (unsigned)W);   // magic-multiply div by 960
        const int   py  = (int)(p / (unsigned)W);
        const float fx  = dx + (float)px;
        const float fy  = dy + (float)py;
        const float x0f = floorf(fx);
        const float y0f = floorf(fy);
        const float ax  = fx - x0f;                 // in [0,1)
        const float ay  = fy - y0f;
        const int   x0  = (int)x0f;
        const int   y0  = (int)y0f;
        const int   x1  = x0 + 1;
        const int   y1  = y0 + 1;
        const float wx0 = 1.0f - ax, wx1 = ax;
        const float wy0 = 1.0f - ay, wy1 = ay;

        const bool vx0 = (x0 >= 0) && (x0 < W);
        const bool vx1 = (x1 >= 0) && (x1 < W);
        const bool vy0 = (y0 >= 0) && (y0 < H);
        const bool vy1 = (y1 >= 0) && (y1 < H);

        if (vy0) {
            const size_t row = (size_t)y0 * W;
            if (vx0) atomicAdd(plane + row + x0, val * (wx0 * wy0));
            if (vx1) atomicAdd(plane + row + x1, val * (wx1 * wy0));
        }
        if (vy1) {
            const size_t row = (size_t)y1 * W;
            if (vx0) atomicAdd(plane + row + x0, val * (wx0 * wy1));
            if (vx1) atomicAdd(plane + row + x1, val * (wx1 * wy1));
        }
    }
}

// ---------------------------------------------------------------------------
// Kernel 3: per-pixel guarded reciprocal of the denominator (2.09 M divides
// instead of 73.1 M in the normalize pass).
// ---------------------------------------------------------------------------
__global__ void softsplat_recip(const float* __restrict__ den,
                                float* __restrict__ rden) {
    const unsigned stride = gridDim.x * blockDim.x;
    for (unsigned j = blockIdx.x * blockDim.x + threadIdx.x; j < N_DEN; j += stride) {
        const float d = den[j];
        rden[j] = (fabsf(d) > SPLAT_EPS) ? (1.0f / d) : 1.0f;
    }
}

// ---------------------------------------------------------------------------
// Kernel 4: vectorized normalize. blockIdx.y = b*C + c; b128 load/store of
// out, b128 load of the L2-resident rden plane, 4 multiplies.
// ---------------------------------------------------------------------------
__global__ void softsplat_normalize(float* __restrict__ out,
                                    const float* __restrict__ rden) {
    const unsigned z = blockIdx.y;            // b*C + c
    const unsigned b = z / (unsigned)C;
    float4* __restrict__ o4 = (float4*)(out + (size_t)z * HW);
    const float4* __restrict__ r4 = (const float4*)(rden + (size_t)b * HW);

    const unsigned stride = gridDim.x * blockDim.x;
    for (unsigned j = blockIdx.x * blockDim.x + threadIdx.x; j < HW4; j += stride) {
        float4 o = o4[j];
        const float4 r = r4[j];
        o.x *= r.x; o.y *= r.y; o.z *= r.z; o.w *= r.w;
        o4[j] = o;
    }
}

// ---------------------------------------------------------------------------
// Host entry. Everything on `stream`; no allocation / sync (graph-capturable).
// Workspace layout: den = d_ws[0 .. B*HW), rden = d_ws[B*HW .. 2*B*HW)
// (16.7 MB total).
// ---------------------------------------------------------------------------
extern "C" void kernel_launch(void* const* d_in, const int* in_sizes, int n_in,
                              void* d_out, int out_size, void* d_ws, size_t ws_size,
                              hipStream_t stream) {
    (void)in_sizes; (void)n_in; (void)out_size; (void)ws_size;

    const float* x      = (const float*)d_in[0];   // [B,C,H,W]
    const float* flow   = (const float*)d_in[1];   // [B,2,H,W]
    const float* metric = (const float*)d_in[2];   // [B,1,H,W]
    float*       out    = (float*)d_out;           // [B,C,H,W]
    float*       den    = (float*)d_ws;            // [B,H,W]
    float*       rden   = den + (size_t)N_DEN;     // [B,H,W]

    const int threads = 256;  // 8 wave32s per workgroup

    // Zero pass: b128 streaming stores.
    softsplat_zero<<<32768, threads, 0, stream>>>((float4*)out, (float4*)den);

    // Scatter pass: 144 planes x 510 pixel-blocks; each thread runs ~4
    // grid-stride iterations so the x-prefetch has a target.
    softsplat_scatter<<<dim3(510, (unsigned)B * K), threads, 0, stream>>>(
        x, flow, metric, out, den);

    // Guarded reciprocal of the denominator plane.
    softsplat_recip<<<2048, threads, 0, stream>>>(den, rden);

    // Normalize pass: b128 in/out, multiply by L2-resident reciprocal.
    softsplat_normalize<<<dim3(510, (unsigned)B * C), threads, 0, stream>>>(out, rden);
}